// TopologyModule_80470507258425
// MI455X (gfx1250) — compile-verified
//
#include <hip/hip_runtime.h>
#include <math.h>

// Cumulative max along channel axis for x[B=2, C=128, N=250000] (f32).
// Memory-bound streaming scan: 256 MB in + 256 MB out -> ~22 us @ 23.3 TB/s.
// One thread per float4 column; run-max in VGPRs initialized to -inf so all
// 128 channels process in 16 batches of 8. Each batch issues 8 independent
// NT b128 loads (4 KB in flight per wave) before the dependent max/store
// chain, plus global_prefetch_b8 two batches ahead (each channel step is a
// separate cacheline: 1 MB stride).

typedef float v4f __attribute__((ext_vector_type(4)));

constexpr int  kB = 2;
constexpr int  kC = 128;
constexpr long kN = 250000;
constexpr int  kN4 = (int)(kN / 4);       // 62500 float4 columns per batch dim
constexpr int  kTotal = kB * kN4;         // 125000 threads
constexpr long kS = kN / 4;               // v4f stride between channels

__device__ __forceinline__ v4f vmax4(v4f a, v4f b) {
    v4f r;
    r.x = fmaxf(a.x, b.x);
    r.y = fmaxf(a.y, b.y);
    r.z = fmaxf(a.z, b.z);
    r.w = fmaxf(a.w, b.w);
    return r;
}

__global__ __launch_bounds__(256)
void topo_cummax_kernel(const float* __restrict__ x, float* __restrict__ out) {
    int tid = blockIdx.x * blockDim.x + threadIdx.x;
    if (tid >= kTotal) return;

    int b  = tid / kN4;
    int n4 = tid - b * kN4;

    const v4f* __restrict__ src = (const v4f*)(x   + (size_t)b * kC * kN) + n4;
    v4f*       __restrict__ dst = (v4f*)      (out + (size_t)b * kC * kN) + n4;

    v4f run;
    run.x = -INFINITY; run.y = -INFINITY; run.z = -INFINITY; run.w = -INFINITY;

    for (int c = 0; c < kC; c += 8) {
        // Prefetch the batch two ahead; each channel is its own cacheline.
        if (c + 16 < kC) {
#pragma unroll
            for (int p = 0; p < 8; ++p) {
                __builtin_prefetch((const void*)(src + (size_t)(c + 16 + p) * kS), 0, 1);
            }
        }

        // 8 independent NT loads issued before any dependent use.
        v4f v0 = __builtin_nontemporal_load(src + (size_t)(c + 0) * kS);
        v4f v1 = __builtin_nontemporal_load(src + (size_t)(c + 1) * kS);
        v4f v2 = __builtin_nontemporal_load(src + (size_t)(c + 2) * kS);
        v4f v3 = __builtin_nontemporal_load(src + (size_t)(c + 3) * kS);
        v4f v4 = __builtin_nontemporal_load(src + (size_t)(c + 4) * kS);
        v4f v5 = __builtin_nontemporal_load(src + (size_t)(c + 5) * kS);
        v4f v6 = __builtin_nontemporal_load(src + (size_t)(c + 6) * kS);
        v4f v7 = __builtin_nontemporal_load(src + (size_t)(c + 7) * kS);

        // Sequential scan chain; stores drain on STOREcnt, independent of
        // the next batch's loads.
        run = vmax4(run, v0); __builtin_nontemporal_store(run, dst + (size_t)(c + 0) * kS);
        run = vmax4(run, v1); __builtin_nontemporal_store(run, dst + (size_t)(c + 1) * kS);
        run = vmax4(run, v2); __builtin_nontemporal_store(run, dst + (size_t)(c + 2) * kS);
        run = vmax4(run, v3); __builtin_nontemporal_store(run, dst + (size_t)(c + 3) * kS);
        run = vmax4(run, v4); __builtin_nontemporal_store(run, dst + (size_t)(c + 4) * kS);
        run = vmax4(run, v5); __builtin_nontemporal_store(run, dst + (size_t)(c + 5) * kS);
        run = vmax4(run, v6); __builtin_nontemporal_store(run, dst + (size_t)(c + 6) * kS);
        run = vmax4(run, v7); __builtin_nontemporal_store(run, dst + (size_t)(c + 7) * kS);
    }
}

extern "C" void kernel_launch(void* const* d_in, const int* in_sizes, int n_in,
                              void* d_out, int out_size, void* d_ws, size_t ws_size,
                              hipStream_t stream) {
    (void)in_sizes; (void)n_in; (void)out_size; (void)d_ws; (void)ws_size;
    const float* x = (const float*)d_in[0];
    float* out = (float*)d_out;

    const int threads = 256;                              // 8 wave32s per block
    const int blocks = (kTotal + threads - 1) / threads;  // 489 blocks
    topo_cummax_kernel<<<blocks, threads, 0, stream>>>(x, out);
}